// ConvFilterNorm_9517647528372
// MI455X (gfx1250) — compile-verified
//
#include <hip/hip_runtime.h>
#include <hip/hip_bf16.h>

// Spectral-norm power iteration for conv filter [2048,2048,3,3] viewed as
// W[6144,6144] with W[o*3+r, i*3+c] = F[o,i,r,c].
// 21 passes over the 151MB filter; filter stays resident in MI455X's 192MB L2
// after the first pass -> L2-bandwidth bound. Row pass uses V_WMMA_F32_16X16X4_F32.

typedef __attribute__((ext_vector_type(2))) float v2f;
typedef __attribute__((ext_vector_type(8))) float v8f;

#define OC   2048
#define NDIM 6144          // = out_ch*3 = in_ch*3
#define KT   768           // K tile per LDS stage
#define PAD  772           // padded LDS row stride (772 % 64 == 4 -> conflict-free)

// ---------------------------------------------------------------------------
// v[m] = sum_n W[m,n] * u[n]   (row pass, WMMA f32 16x16x4)
// One 16-row block per 256-thread WG (8 waves). K tiled through LDS; wave w
// owns K sub-stripe [w*96, w*96+96) of each tile; cross-wave reduce at end.
// B is the u-chunk replicated across all 16 N columns, so every D column
// equals the matvec partial (lane0/vgpr j -> row j, lane16/vgpr j -> row j+8).
// ---------------------------------------------------------------------------
__global__ __launch_bounds__(256) void matvec_rows_wmma(
    const float* __restrict__ F, const float* __restrict__ u, float* __restrict__ v)
{
    __shared__ float lA[16 * PAD];   // 49,408 B
    __shared__ float red[8][16];

    const int tid  = threadIdx.x;
    const int wave = tid >> 5;
    const int lane = tid & 31;
    const int m0   = blockIdx.x * 16;

    v8f acc = {};

    for (int k0 = 0; k0 < NDIM; k0 += KT) {
        // Cooperative, coalesced stage of W[m0..m0+15, k0..k0+KT) from the
        // permuted filter layout. 16*KT/256 == 48 iterations, no remainder.
        for (int e = tid; e < 16 * KT; e += 256) {
            int rr = e / KT;
            int kk = e - rr * KT;
            int m  = m0 + rr;
            int o  = m / 3, r = m - 3 * o;
            int n  = k0 + kk;
            int i  = n / 3, c = n - 3 * i;
            lA[rr * PAD + kk] = F[(size_t)(o * OC + i) * 9 + r * 3 + c];
        }
        __syncthreads();

        const int ml = lane & 15;              // A/B row within tile
        const int kh = (lane >> 4) << 1;       // K half: 0 (lanes 0-15) / 2 (16-31)
        const int ktbase = wave * (KT / 8);    // 96-wide stripe per wave
        #pragma unroll 4
        for (int kt = ktbase; kt < ktbase + KT / 8; kt += 4) {
            v2f a, b;
            a.x = lA[ml * PAD + kt + kh];
            a.y = lA[ml * PAD + kt + kh + 1];
            const int kg = k0 + kt + kh;
            b.x = u[kg];
            b.y = u[kg + 1];
            acc = __builtin_amdgcn_wmma_f32_16x16x4_f32(
                false, a, false, b, (short)0, acc, false, false);
        }
        __syncthreads();
    }

    // D layout: vgpr j -> M=j (lanes 0-15), M=j+8 (lanes 16-31); all N equal.
    if (lane == 0) {
        #pragma unroll
        for (int j = 0; j < 8; ++j) red[wave][j] = acc[j];
    } else if (lane == 16) {
        #pragma unroll
        for (int j = 0; j < 8; ++j) red[wave][8 + j] = acc[j];
    }
    __syncthreads();
    if (tid < 16) {
        float s = 0.f;
        #pragma unroll
        for (int w = 0; w < 8; ++w) s += red[w][tid];
        v[m0 + tid] = s;
    }
}

// ---------------------------------------------------------------------------
// Transpose pass: part[ob][i*3+c] = sum_{o in ob-block, r} F[o,i,r,c]*vn[o*3+r]
// Deterministic partial sums (no float atomics). Each thread owns one i and
// reads 9 contiguous floats per o (36B) -> fully used cachelines.
// grid (8, 64), block 256.
// ---------------------------------------------------------------------------
__global__ __launch_bounds__(256) void matvec_cols_partial(
    const float* __restrict__ F, const float* __restrict__ vn, float* __restrict__ part)
{
    __shared__ float lv[96];
    const int t  = threadIdx.x;
    const int ib = blockIdx.x;     // 0..7
    const int ob = blockIdx.y;     // 0..63
    const int i  = ib * 256 + t;   // 0..2047

    if (t < 96) lv[t] = vn[ob * 96 + t];
    __syncthreads();

    float a0 = 0.f, a1 = 0.f, a2 = 0.f;
    for (int oo = 0; oo < 32; ++oo) {
        const int o = ob * 32 + oo;
        const float* fp = F + (size_t)(o * OC + i) * 9;
        #pragma unroll
        for (int r = 0; r < 3; ++r) {
            const float vr = lv[oo * 3 + r];
            a0 = fmaf(fp[r * 3 + 0], vr, a0);
            a1 = fmaf(fp[r * 3 + 1], vr, a1);
            a2 = fmaf(fp[r * 3 + 2], vr, a2);
        }
    }
    float* p = part + (size_t)ob * NDIM + i * 3;
    p[0] = a0; p[1] = a1; p[2] = a2;
}

__global__ __launch_bounds__(256) void reduce_partials(
    const float* __restrict__ part, float* __restrict__ out)
{
    const int n = blockIdx.x * 256 + threadIdx.x;   // grid 24 -> 6144
    float s = 0.f;
    #pragma unroll 8
    for (int b = 0; b < 64; ++b) s += part[(size_t)b * NDIM + n];
    out[n] = s;
}

// x <- x / ||x||  (single WG; 6144 elements is trivial)
__global__ __launch_bounds__(256) void normalize_vec(float* __restrict__ x)
{
    __shared__ float red[256];
    __shared__ float inv;
    float s = 0.f;
    for (int idx = threadIdx.x; idx < NDIM; idx += 256) { float t = x[idx]; s += t * t; }
    red[threadIdx.x] = s;
    __syncthreads();
    for (int off = 128; off > 0; off >>= 1) {
        if (threadIdx.x < off) red[threadIdx.x] += red[threadIdx.x + off];
        __syncthreads();
    }
    if (threadIdx.x == 0) inv = 1.0f / sqrtf(red[0]);
    __syncthreads();
    const float iv = inv;
    for (int idx = threadIdx.x; idx < NDIM; idx += 256) x[idx] *= iv;
}

// out[0] = sqrt(h*w) * dot(a, b) = 3 * dot(a, b)
__global__ __launch_bounds__(256) void sigma_dot(
    const float* __restrict__ a, const float* __restrict__ b, float* __restrict__ out)
{
    __shared__ float red[256];
    float s = 0.f;
    for (int idx = threadIdx.x; idx < NDIM; idx += 256) s += a[idx] * b[idx];
    red[threadIdx.x] = s;
    __syncthreads();
    for (int off = 128; off > 0; off >>= 1) {
        if (threadIdx.x < off) red[threadIdx.x] += red[threadIdx.x + off];
        __syncthreads();
    }
    if (threadIdx.x == 0) out[0] = 3.0f * red[0];
}

__global__ __launch_bounds__(256) void copy_vec(
    const float* __restrict__ in, float* __restrict__ out)
{
    const int idx = blockIdx.x * 256 + threadIdx.x;
    if (idx < NDIM) out[idx] = in[idx];
}

// ---------------------------------------------------------------------------
extern "C" void kernel_launch(void* const* d_in, const int* in_sizes, int n_in,
                              void* d_out, int out_size, void* d_ws, size_t ws_size,
                              hipStream_t stream)
{
    const float* F    = (const float*)d_in[0];   // [2048,2048,3,3] fp32
    const float* u_in = (const float*)d_in[1];   // [1,6144] fp32, unit norm
    float* out = (float*)d_out;

    float* ws   = (float*)d_ws;                  // needs (3+64)*6144*4 ~ 1.65 MB
    float* u    = ws;                            // current u  (6144)
    float* v    = ws + NDIM;                     // current v  (6144)
    float* w    = ws + 2 * NDIM;                 // W @ u_n    (6144)
    float* part = ws + 3 * NDIM;                 // 64 x 6144 partials

    copy_vec<<<24, 256, 0, stream>>>(u_in, u);   // never mutate inputs

    const dim3 colGrid(8, 64);
    for (int it = 0; it < 10; ++it) {
        matvec_rows_wmma<<<NDIM / 16, 256, 0, stream>>>(F, u, v);   // v = W u
        normalize_vec<<<1, 256, 0, stream>>>(v);                    // v_n
        matvec_cols_partial<<<colGrid, 256, 0, stream>>>(F, v, part); // Wt v_n
        reduce_partials<<<24, 256, 0, stream>>>(part, u);           // u2
        normalize_vec<<<1, 256, 0, stream>>>(u);                    // u_n
    }
    matvec_rows_wmma<<<NDIM / 16, 256, 0, stream>>>(F, u, w);       // w = W u_n
    sigma_dot<<<1, 256, 0, stream>>>(v, w, out);                    // 3 * v_n.w
}